// HGT_49065706389937
// MI455X (gfx1250) — compile-verified
//
#include <hip/hip_runtime.h>
#include <math.h>

typedef __attribute__((ext_vector_type(2))) float v2f;
typedef __attribute__((ext_vector_type(8))) float v8f;

static constexpr int NA_   = 100000;
static constexpr int NB_   = 100000;
static constexpr int NT_   = NA_ + NB_;
static constexpr int FEAT_ = 128;
static constexpr int H_    = 4;
static constexpr int EAB_  = 500000;
static constexpr int EBA_  = 500000;
static constexpr int ETOT_ = EAB_ + EBA_;

// ---- order-preserving float <-> uint mapping (for atomic segment-max) ----
__device__ __forceinline__ unsigned f2mono(float x) {
  unsigned u = __float_as_uint(x);
  return (u & 0x80000000u) ? ~u : (u | 0x80000000u);
}
__device__ __forceinline__ float mono2f(unsigned m) {
  unsigned u = (m & 0x80000000u) ? (m & 0x7FFFFFFFu) : ~m;
  return __uint_as_float(u);
}

// =====================================================================
// 0) Fuse per-head relation transforms into the KQV weights.
//    Wf[t] is [128,384]:  cols [0,128)  = W_k · blockdiag(W_krel[h*2+t])
//                         cols [128,256)= W_q (copied)
//                         cols [256,384)= W_v · blockdiag(W_vrel[h*2+t])
//    bf[t][384] gets the same transform applied to b_kqv.
// =====================================================================
__global__ void fuse_weights(const float* __restrict__ W_kqv, const float* __restrict__ b_kqv,
                             const float* __restrict__ W_krel, const float* __restrict__ W_vrel,
                             float* __restrict__ Wf, float* __restrict__ bf) {
  int idx = blockIdx.x * blockDim.x + threadIdx.x;
  const int total = 2 * 129 * 384;            // i==128 row computes the bias
  if (idx >= total) return;
  int t = idx / (129 * 384);
  int r = idx % (129 * 384);
  int i = r / 384;
  int j = r % 384;
  const float* Wt = W_kqv + t * 128 * 384;
  const float* bt = b_kqv + t * 384;
  float val;
  if (j >= 128 && j < 256) {
    val = (i < 128) ? Wt[i * 384 + j] : bt[j];
  } else {
    int part = (j < 128) ? 0 : 2;             // k part or v part
    int jj   = (j < 128) ? j : (j - 256);
    int h = jj >> 5, jd = jj & 31;
    const float* Wrel = ((j < 128) ? W_krel : W_vrel) + (size_t)(h * 2 + t) * 32 * 32;
    float acc = 0.f;
#pragma unroll
    for (int d = 0; d < 32; ++d) {
      float s = (i < 128) ? Wt[i * 384 + part * 128 + h * 32 + d]
                          : bt[part * 128 + h * 32 + d];
      acc += s * Wrel[d * 32 + jd];
    }
    val = acc;
  }
  if (i < 128) Wf[(size_t)t * 128 * 384 + i * 384 + j] = val;
  else         bf[t * 384 + j] = val;
}

// =====================================================================
// 1) KQV GEMM with fused rel-transform: Y[N,384] = X[N,128] @ Wf + bf
//    BM=64 rows/block, 8 waves; wave w owns N-tiles {3w,3w+1,3w+2}, all 4
//    16-row subtiles. f32 WMMA: 16x16x4, K-loop of 32.
// =====================================================================
__global__ __launch_bounds__(256) void gemm_kqv(
    const float* __restrict__ X, int N,
    const float* __restrict__ Wf, const float* __restrict__ bf,
    float* __restrict__ Kout, float* __restrict__ Qout, float* __restrict__ Vout,
    int rowOff) {
  __shared__ float Atile[64 * 128];
  const int tid  = threadIdx.x;
  const int wave = tid >> 5, lane = tid & 31;
  const int half = lane >> 4, l16 = lane & 15;
  const int blockRow = blockIdx.x * 64;

  // stage A tile (64x128 f32) into LDS, zero-padded past N
#pragma unroll
  for (int iLoad = 0; iLoad < 8; ++iLoad) {
    int idx = tid + iLoad * 256;            // float4 slot id, 2048 total
    int r = idx >> 5, c4 = idx & 31;
    float4 v = make_float4(0.f, 0.f, 0.f, 0.f);
    int gr = blockRow + r;
    if (gr < N) v = *(const float4*)(X + (size_t)gr * 128 + c4 * 4);
    *(float4*)(Atile + r * 128 + c4 * 4) = v;
  }
  __syncthreads();

  v8f acc[4][3] = {};
  for (int it = 0; it < 32; ++it) {
    const int kk = it * 4 + half * 2;       // this lane's K pair
    v2f bfrag[3];
#pragma unroll
    for (int jj = 0; jj < 3; ++jj) {
      int n = (wave * 3 + jj) * 16 + l16;
      bfrag[jj].x = Wf[(size_t)kk * 384 + n];
      bfrag[jj].y = Wf[(size_t)(kk + 1) * 384 + n];
    }
#pragma unroll
    for (int r = 0; r < 4; ++r) {
      v2f afrag = *(const v2f*)(Atile + (r * 16 + l16) * 128 + kk);
#pragma unroll
      for (int jj = 0; jj < 3; ++jj)
        acc[r][jj] = __builtin_amdgcn_wmma_f32_16x16x4_f32(
            false, afrag, false, bfrag[jj], (short)0, acc[r][jj], false, false);
    }
  }

  // epilogue: bias + scatter into k_src / q_cat / v_src
#pragma unroll
  for (int r = 0; r < 4; ++r) {
#pragma unroll
    for (int jj = 0; jj < 3; ++jj) {
      int c = (wave * 3 + jj) * 16 + l16;   // output column in [0,384)
      float bias = bf[c];
      float* dstp; int cc;
      if (c < 128)      { dstp = Kout; cc = c; }
      else if (c < 256) { dstp = Qout; cc = c - 128; }
      else              { dstp = Vout; cc = c - 256; }
#pragma unroll
      for (int i = 0; i < 8; ++i) {
        int m = r * 16 + i + half * 8;
        int gr = blockRow + m;
        if (gr < N)
          dstp[(size_t)(rowOff + gr) * 128 + cc] = acc[r][jj][i] + bias;
      }
    }
  }
}

// ---- edge decode helper: combined [a->b ; b->a] edge space ----
__device__ __forceinline__ void edge_decode(int e,
    const int* __restrict__ ab_s, const int* __restrict__ ab_d,
    const int* __restrict__ ba_s, const int* __restrict__ ba_d,
    int& src, int& dst, int& et) {
  if (e < EAB_) { src = ab_s[e];        dst = ab_d[e] + NA_; et = 0; }
  else { int k = e - EAB_; src = ba_s[k] + NA_; dst = ba_d[k]; et = 1; }
}

// =====================================================================
// 2a) alpha[e,h] = (q[dst]·k[src])_h * p_rel / sqrt(D);  atomicMax segment max
// =====================================================================
__global__ __launch_bounds__(256) void edge_alpha(
    const int* __restrict__ ab_s, const int* __restrict__ ab_d,
    const int* __restrict__ ba_s, const int* __restrict__ ba_d,
    const float* __restrict__ Ksrc, const float* __restrict__ Q,
    const float* __restrict__ p_rel,
    float* __restrict__ alpha, unsigned* __restrict__ amax) {
  int e = (blockIdx.x * blockDim.x + threadIdx.x) >> 5;
  int lane = threadIdx.x & 31;
  if (e >= ETOT_) return;
  int src, dst, et;
  edge_decode(e, ab_s, ab_d, ba_s, ba_d, src, dst, et);
  float4 qv = *(const float4*)(Q    + (size_t)dst * 128 + lane * 4);
  float4 kv = *(const float4*)(Ksrc + (size_t)src * 128 + lane * 4);
  float p = qv.x * kv.x + qv.y * kv.y + qv.z * kv.z + qv.w * kv.w;
  p += __shfl_xor(p, 1, 32);
  p += __shfl_xor(p, 2, 32);
  p += __shfl_xor(p, 4, 32);                 // sum within 8-lane head group
  int h = lane >> 3;
  if ((lane & 7) == 0) {
    float a = p * p_rel[et * H_ + h] * 0.17677669529663687f;  // 1/sqrt(32)
    alpha[(size_t)e * H_ + h] = a;
    atomicMax(&amax[(size_t)dst * H_ + h], f2mono(a));
  }
}

// =====================================================================
// 2b) aexp = exp(alpha - amax[dst]); atomic segment sum; alpha <- aexp
// =====================================================================
__global__ __launch_bounds__(256) void edge_expsum(
    const int* __restrict__ ab_s, const int* __restrict__ ab_d,
    const int* __restrict__ ba_s, const int* __restrict__ ba_d,
    float* __restrict__ alpha, const unsigned* __restrict__ amax,
    float* __restrict__ asum) {
  int gid = blockIdx.x * blockDim.x + threadIdx.x;
  int e = gid >> 2, h = gid & 3;
  if (e >= ETOT_) return;
  int src, dst, et;
  edge_decode(e, ab_s, ab_d, ba_s, ba_d, src, dst, et);
  float a  = alpha[(size_t)e * H_ + h];
  float ex = expf(a - mono2f(amax[(size_t)dst * H_ + h]));
  alpha[(size_t)e * H_ + h] = ex;
  atomicAdd(&asum[(size_t)dst * H_ + h], ex);
}

// =====================================================================
// 2c) attn = aexp / (asum + 1e-16); aggr[dst] += attn * v[src]
// =====================================================================
__global__ __launch_bounds__(256) void edge_scatter(
    const int* __restrict__ ab_s, const int* __restrict__ ab_d,
    const int* __restrict__ ba_s, const int* __restrict__ ba_d,
    const float* __restrict__ Vsrc, const float* __restrict__ alpha,
    const float* __restrict__ asum, float* __restrict__ aggr) {
  int e = (blockIdx.x * blockDim.x + threadIdx.x) >> 5;
  int lane = threadIdx.x & 31;
  if (e >= ETOT_) return;
  int src, dst, et;
  edge_decode(e, ab_s, ab_d, ba_s, ba_d, src, dst, et);
  int h = lane >> 3;
  float attn = alpha[(size_t)e * H_ + h] / (asum[(size_t)dst * H_ + h] + 1e-16f);
  float4 vv = *(const float4*)(Vsrc + (size_t)src * 128 + lane * 4);
  float* o = aggr + (size_t)dst * 128 + lane * 4;
  atomicAdd(o + 0, vv.x * attn);
  atomicAdd(o + 1, vv.y * attn);
  atomicAdd(o + 2, vv.z * attn);
  atomicAdd(o + 3, vv.w * attn);
}

// =====================================================================
// 3) finish: res = g·(gelu(aggr)@W_out + b_out) + (1-g)·x ; logits = res@W_cls
//    16 rows/block; wave w computes N-tile w via f32 WMMA; classifier from LDS.
// =====================================================================
__global__ __launch_bounds__(256) void finish(
    const float* __restrict__ aggr, const float* __restrict__ X,
    const float* __restrict__ Wout, const float* __restrict__ bout,
    const float* __restrict__ skip,
    const float* __restrict__ Wcls, const float* __restrict__ bcls,
    float* __restrict__ logits, int N, int rowOff) {
  __shared__ float Atile[16 * 128];
  __shared__ float Res[16 * 128];
  const int tid = threadIdx.x;
  const int wave = tid >> 5, lane = tid & 31;
  const int half = lane >> 4, l16 = lane & 15;
  const int blockRow = blockIdx.x * 16;

  // load + exact GELU into LDS
#pragma unroll
  for (int i = 0; i < 8; ++i) {
    int idx = tid + i * 256;
    int r = idx >> 7, c = idx & 127;
    int gr = blockRow + r;
    float x = (gr < N) ? aggr[(size_t)(rowOff + gr) * 128 + c] : 0.f;
    Atile[idx] = 0.5f * x * (1.f + erff(x * 0.70710678118654752f));
  }
  __syncthreads();

  v8f acc = {};
  for (int it = 0; it < 32; ++it) {
    int kk = it * 4 + half * 2;
    v2f afrag = *(const v2f*)(Atile + l16 * 128 + kk);
    v2f bfrag;
    bfrag.x = Wout[(size_t)kk * 128 + wave * 16 + l16];
    bfrag.y = Wout[(size_t)(kk + 1) * 128 + wave * 16 + l16];
    acc = __builtin_amdgcn_wmma_f32_16x16x4_f32(
        false, afrag, false, bfrag, (short)0, acc, false, false);
  }

  float g = 1.f / (1.f + expf(-skip[0]));
  int c = wave * 16 + l16;
  float bo = bout[c];
#pragma unroll
  for (int i = 0; i < 8; ++i) {
    int m = i + half * 8;
    int gr = blockRow + m;
    float res = 0.f;
    if (gr < N) {
      float a = acc[i] + bo;
      res = g * a + (1.f - g) * X[(size_t)gr * 128 + c];
    }
    Res[m * 128 + c] = res;
  }
  __syncthreads();

  if (tid < 48) {                          // 16 rows x 3 classes
    int r = tid / 3, cc = tid % 3;
    int gr = blockRow + r;
    if (gr < N) {
      float s = bcls[cc];
      for (int k = 0; k < 128; ++k) s += Res[r * 128 + k] * Wcls[k * 3 + cc];
      logits[(size_t)(rowOff + gr) * 3 + cc] = s;
    }
  }
}

// =====================================================================
extern "C" void kernel_launch(void* const* d_in, const int* in_sizes, int n_in,
                              void* d_out, int out_size, void* d_ws, size_t ws_size,
                              hipStream_t stream) {
  const float* x_a    = (const float*)d_in[0];
  const float* x_b    = (const float*)d_in[1];
  const float* W_kqv  = (const float*)d_in[2];
  const float* b_kqv  = (const float*)d_in[3];
  const float* W_krel = (const float*)d_in[4];
  const float* W_vrel = (const float*)d_in[5];
  const float* p_rel  = (const float*)d_in[6];
  const float* W_out  = (const float*)d_in[7];
  const float* b_out  = (const float*)d_in[8];
  const float* skip   = (const float*)d_in[9];
  const float* W_cls  = (const float*)d_in[10];
  const float* b_cls  = (const float*)d_in[11];
  const int* edge_ab  = (const int*)d_in[12];
  const int* edge_ba  = (const int*)d_in[13];
  float* logits = (float*)d_out;
  (void)in_sizes; (void)n_in; (void)out_size; (void)ws_size;

  char* ws = (char*)d_ws;
  auto alloc = [&](size_t bytes) -> char* {
    char* p = ws; ws += (bytes + 255) & ~(size_t)255; return p;
  };
  float*    Wf    = (float*)   alloc((size_t)2 * 128 * 384 * 4);
  float*    bfb   = (float*)   alloc((size_t)2 * 384 * 4);
  float*    Ksrc  = (float*)   alloc((size_t)NT_ * FEAT_ * 4);
  float*    Q     = (float*)   alloc((size_t)NT_ * FEAT_ * 4);
  float*    Vsrc  = (float*)   alloc((size_t)NT_ * FEAT_ * 4);
  float*    aggr  = (float*)   alloc((size_t)NT_ * FEAT_ * 4);
  float*    alpha = (float*)   alloc((size_t)ETOT_ * H_ * 4);
  unsigned* amax  = (unsigned*)alloc((size_t)NT_ * H_ * 4);
  float*    asum  = (float*)   alloc((size_t)NT_ * H_ * 4);

  hipMemsetAsync(aggr, 0, (size_t)NT_ * FEAT_ * 4, stream);
  hipMemsetAsync(amax, 0, (size_t)NT_ * H_ * 4, stream);   // mono(0) == -inf
  hipMemsetAsync(asum, 0, (size_t)NT_ * H_ * 4, stream);

  {
    int total = 2 * 129 * 384;
    fuse_weights<<<(total + 255) / 256, 256, 0, stream>>>(
        W_kqv, b_kqv, W_krel, W_vrel, Wf, bfb);
  }

  gemm_kqv<<<(NA_ + 63) / 64, 256, 0, stream>>>(x_a, NA_, Wf, bfb, Ksrc, Q, Vsrc, 0);
  gemm_kqv<<<(NB_ + 63) / 64, 256, 0, stream>>>(x_b, NB_, Wf + 128 * 384, bfb + 384,
                                                Ksrc, Q, Vsrc, NA_);

  const int* ab_s = edge_ab;        const int* ab_d = edge_ab + EAB_;
  const int* ba_s = edge_ba;        const int* ba_d = edge_ba + EBA_;

  edge_alpha<<<(ETOT_ + 7) / 8, 256, 0, stream>>>(ab_s, ab_d, ba_s, ba_d,
                                                  Ksrc, Q, p_rel, alpha, amax);
  edge_expsum<<<((ETOT_ * H_) + 255) / 256, 256, 0, stream>>>(ab_s, ab_d, ba_s, ba_d,
                                                              alpha, amax, asum);
  edge_scatter<<<(ETOT_ + 7) / 8, 256, 0, stream>>>(ab_s, ab_d, ba_s, ba_d,
                                                    Vsrc, alpha, asum, aggr);

  finish<<<(NA_ + 15) / 16, 256, 0, stream>>>(aggr, x_a, W_out, b_out, skip,
                                              W_cls, b_cls, logits, NA_, 0);
  finish<<<(NB_ + 15) / 16, 256, 0, stream>>>(aggr, x_b, W_out + 128 * 128,
                                              b_out + 128, skip + 1,
                                              W_cls, b_cls, logits, NB_, NA_);
}